// ScalableHashGrid_24721831756642
// MI455X (gfx1250) — compile-verified
//
#include <hip/hip_runtime.h>

typedef float v4f __attribute__((ext_vector_type(4)));

#define NPTS   (1u << 20)
#define NLEV   10
#define HMASK  ((1u << 21) - 1u)

// Row offsets (in table rows of 4 floats) per level, from the reference's
// padded cumulative sizes: dense 17^3/33^3/65^3 (padded to 8), then 7 x 2^21.
__constant__ unsigned kOff[NLEV] = {
    0u, 4920u, 40864u, 315496u, 2412648u, 4509800u,
    6606952u, 8704104u, 10801256u, 12898408u};

__global__ __launch_bounds__(256, 2) void hashgrid_kernel(
    const float* __restrict__ xin, const v4f* __restrict__ table,
    float* __restrict__ out)
{
  const unsigned tid   = threadIdx.x;
  const unsigned level = blockIdx.y;
  const unsigned n     = blockIdx.x * 256u + tid;

  // Stage 256 points' coords through LDS with fully coalesced global loads.
  __shared__ float xs[256 * 3];
  {
    const float* xg = xin + (size_t)blockIdx.x * (256u * 3u);
    xs[tid]        = xg[tid];
    xs[tid + 256u] = xg[tid + 256u];
    xs[tid + 512u] = xg[tid + 512u];
  }
  __syncthreads();

  // stride-3 LDS read: 3*i mod 64 is bank-conflict free (gcd(3,64)=1)
  const float res_m1 = (float)((16u << level) - 1u);
  const float px = ((xs[tid * 3u + 0u] + 1.0f) * 0.5f) * res_m1 + 0.5f;
  const float py = ((xs[tid * 3u + 1u] + 1.0f) * 0.5f) * res_m1 + 0.5f;
  const float pz = ((xs[tid * 3u + 2u] + 1.0f) * 0.5f) * res_m1 + 0.5f;

  const float bx = floorf(px), by = floorf(py), bz = floorf(pz);
  const float fx = px - bx, fy = py - by, fz = pz - bz;
  const unsigned x0 = (unsigned)bx, y0 = (unsigned)by, z0 = (unsigned)bz;

  // Corner i: x += i&1, y += (i>>1)&1, z += (i>>2)&1  (matches CORNER_OFFS)
  unsigned i0, i1, i2, i3, i4, i5, i6, i7;
  if (level < 3u) {                       // dense levels: (res+1)^3 table
    const unsigned s  = (16u << level) + 1u;
    const unsigned s2 = s * s;
    const unsigned b  = x0 + y0 * s + z0 * s2;
    i0 = b;           i1 = b + 1u;
    i2 = b + s;       i3 = b + s + 1u;
    i4 = b + s2;      i5 = b + s2 + 1u;
    i6 = b + s2 + s;  i7 = b + s2 + s + 1u;
  } else {                                // hashed levels
    const unsigned hy0 = y0 * 2654435761u, hy1 = (y0 + 1u) * 2654435761u;
    const unsigned hz0 = z0 * 805459861u,  hz1 = (z0 + 1u) * 805459861u;
    const unsigned x1  = x0 + 1u;
    i0 = (x0 ^ hy0 ^ hz0) & HMASK;  i1 = (x1 ^ hy0 ^ hz0) & HMASK;
    i2 = (x0 ^ hy1 ^ hz0) & HMASK;  i3 = (x1 ^ hy1 ^ hz0) & HMASK;
    i4 = (x0 ^ hy0 ^ hz1) & HMASK;  i5 = (x1 ^ hy0 ^ hz1) & HMASK;
    i6 = (x0 ^ hy1 ^ hz1) & HMASK;  i7 = (x1 ^ hy1 ^ hz1) & HMASK;
  }

  // Issue all 8 16-byte gathers before consuming -> max outstanding LOADcnt.
  const v4f* __restrict__ tab = table + kOff[level];
  const v4f e0 = tab[i0], e1 = tab[i1], e2 = tab[i2], e3 = tab[i3];
  const v4f e4 = tab[i4], e5 = tab[i5], e6 = tab[i6], e7 = tab[i7];

  const float gx = 1.0f - fx, gy = 1.0f - fy, gz = 1.0f - fz;
  const float w00 = gy * gz, w10 = fy * gz, w01 = gy * fz, w11 = fy * fz;

  v4f acc = e0 * (gx * w00);
  acc += e1 * (fx * w00);
  acc += e2 * (gx * w10);
  acc += e3 * (fx * w10);
  acc += e4 * (gx * w01);
  acc += e5 * (fx * w01);
  acc += e6 * (gx * w11);
  acc += e7 * (fx * w11);

  // Streaming 160MB output: non-temporal 128-bit store so it does not evict
  // the L2-resident hash tables. Byte offset n*160 + level*16 is 16B aligned.
  v4f* op = (v4f*)(out + (size_t)n * 40u + (size_t)level * 4u);
  __builtin_nontemporal_store(acc, op);
}

extern "C" void kernel_launch(void* const* d_in, const int* in_sizes, int n_in,
                              void* d_out, int out_size, void* d_ws, size_t ws_size,
                              hipStream_t stream) {
  const float* x     = (const float*)d_in[0];   // [2^20, 3] f32
  const v4f*   table = (const v4f*)d_in[1];     // [TOTAL, 4] f32 rows
  float*       out   = (float*)d_out;           // [2^20, 40] f32

  dim3 grid(NPTS / 256u, NLEV);
  dim3 block(256);
  hashgrid_kernel<<<grid, block, 0, stream>>>(x, table, out);
}